// GINEncoder_74466142978137
// MI455X (gfx1250) — compile-verified
//
#include <hip/hip_runtime.h>

// GIN encoder, fp32 end-to-end. GEMMs run on V_WMMA_F32_16X16X4_F32 so fp32
// semantics match the JAX reference exactly while using the CDNA5 matrix pipe.
//
// Inputs (setup_inputs order):
//  0 x[100000,3] f32   1 edge_index[2,1600000] i32   2 batch[100000] i32
//  3 W1a[3,64] 4 b1a[64] 5 W1b[64,64] 6 b1b[64]
//  7 W2a[64,64] 8 b2a[64] 9 W2b[64,64] 10 b2b[64]
// Output: [128,64] f32.

typedef __attribute__((ext_vector_type(2))) float v2f;
typedef __attribute__((ext_vector_type(8))) float v8f;

#define N_NODES  100000
#define N_EDGES  1600000
#define N_GRAPHS 128
#define N_TILES  (N_NODES / 16)   // 6250, exact

__global__ void zero_kernel(float* __restrict__ p, int n) {
  int i = blockIdx.x * blockDim.x + threadIdx.x;
  if (i < n) p[i] = 0.0f;
}

// Layer-1 aggregation: agg3[dst] += x[src], 3 floats per edge.
__global__ void scatter3_kernel(const float* __restrict__ x,
                                const int* __restrict__ ei,
                                float* __restrict__ agg3) {
  int e = blockIdx.x * blockDim.x + threadIdx.x;
  if (e >= N_EDGES) return;
  int s = ei[e];
  int d = ei[N_EDGES + e];
  #pragma unroll
  for (int k = 0; k < 3; ++k)
    atomicAdd(&agg3[d * 3 + k], x[s * 3 + k]);
}

// Layer-2 aggregation: 64 threads per edge, coalesced gather + f32 atomics.
__global__ void scatter64_kernel(const float* __restrict__ h,
                                 const int* __restrict__ ei,
                                 float* __restrict__ agg) {
  long long gid = (long long)blockIdx.x * blockDim.x + threadIdx.x;
  int e = (int)(gid >> 6);
  int f = (int)(gid & 63);
  if (e >= N_EDGES) return;
  int s = ei[e];
  int d = ei[N_EDGES + e];
  atomicAdd(&agg[d * 64 + f], h[s * 64 + f]);
}

// MLP layer 1: h1 = relu(relu((x+agg3) @ W1a + b1a) @ W1b + b1b)
// One wave per 16-node tile. K of first GEMM padded 3 -> 4.
__global__ __launch_bounds__(128) void mlp1_kernel(
    const float* __restrict__ x, const float* __restrict__ agg3,
    const float* __restrict__ W1a, const float* __restrict__ b1a,
    const float* __restrict__ W1b, const float* __restrict__ b1b,
    float* __restrict__ h1) {
  __shared__ float sWa[4 * 64];     // W1a padded to K=4 (row 3 = 0)
  __shared__ float sWb[64 * 64];
  __shared__ float sBa[64];
  __shared__ float sBb[64];
  __shared__ float sT[4][16 * 68];  // per-wave 16x64 scratch, row stride 68

  int tid = threadIdx.x;
  for (int i = tid; i < 4 * 64; i += 128) {
    int k = i >> 6, n = i & 63;
    sWa[i] = (k < 3) ? W1a[k * 64 + n] : 0.0f;
  }
  for (int i = tid; i < 64 * 64; i += 128) sWb[i] = W1b[i];
  if (tid < 64) { sBa[tid] = b1a[tid]; sBb[tid] = b1b[tid]; }
  __syncthreads();

  int wave = tid >> 5;
  int lane = tid & 31;
  int half = lane >> 4;   // 0: K pair {k0,k0+1}, 1: {k0+2,k0+3}
  int m    = lane & 15;
  int tile = blockIdx.x * 4 + wave;
  if (tile >= N_TILES) return;               // wave-uniform: EXEC stays all-1
  int node0 = tile * 16;
  float* myT = &sT[wave][0];

  // ---- GEMM1: A = (x+agg3)[16x4pad], B = W1a[4x64]; single K-step ----
  int node = node0 + m;
  int k0   = 2 * half;                        // 0 or 2
  v2f a;
  a.x = x[node * 3 + k0] + agg3[node * 3 + k0];
  a.y = (half == 0) ? (x[node * 3 + 1] + agg3[node * 3 + 1]) : 0.0f;

  v8f acc[4] = {};
  #pragma unroll
  for (int nt = 0; nt < 4; ++nt) {
    v2f b;
    b.x = sWa[k0 * 64 + nt * 16 + m];
    b.y = sWa[(k0 + 1) * 64 + nt * 16 + m];
    acc[nt] = __builtin_amdgcn_wmma_f32_16x16x4_f32(
        false, a, false, b, (short)0, acc[nt], false, false);
  }

  // bias + relu -> stage t (16x64) into LDS (C layout: VGPR r -> rows r, r+8)
  #pragma unroll
  for (int nt = 0; nt < 4; ++nt) {
    float bias = sBa[nt * 16 + m];
    #pragma unroll
    for (int r = 0; r < 8; ++r) {
      float v = acc[nt][r] + bias;
      myT[(r + 8 * half) * 68 + nt * 16 + m] = v > 0.0f ? v : 0.0f;
    }
  }

  // ---- GEMM2: t[16x64] @ W1b[64x64], K=64 -> 16 K-steps x 4 N-tiles ----
  v8f acc2[4] = {};
  for (int ks = 0; ks < 16; ++ks) {
    int kk = ks * 4 + 2 * half;
    v2f a2;
    a2.x = myT[m * 68 + kk];
    a2.y = myT[m * 68 + kk + 1];
    #pragma unroll
    for (int nt = 0; nt < 4; ++nt) {
      v2f b;
      b.x = sWb[kk * 64 + nt * 16 + m];
      b.y = sWb[(kk + 1) * 64 + nt * 16 + m];
      acc2[nt] = __builtin_amdgcn_wmma_f32_16x16x4_f32(
          false, a2, false, b, (short)0, acc2[nt], false, false);
    }
  }

  #pragma unroll
  for (int nt = 0; nt < 4; ++nt) {
    float bias = sBb[nt * 16 + m];
    #pragma unroll
    for (int r = 0; r < 8; ++r) {
      float v = acc2[nt][r] + bias;
      h1[(node0 + r + 8 * half) * 64 + nt * 16 + m] = v > 0.0f ? v : 0.0f;
    }
  }
}

// MLP layer 2: h2 = relu(relu((h1+agg64) @ W2a + b2a) @ W2b + b2b)
// h2 may alias h1: each wave stages its whole input tile to LDS before writing.
__global__ __launch_bounds__(128) void mlp2_kernel(
    const float* __restrict__ h1, const float* __restrict__ agg64,
    const float* __restrict__ W2a, const float* __restrict__ b2a,
    const float* __restrict__ W2b, const float* __restrict__ b2b,
    float* __restrict__ h2) {
  __shared__ float sWa[64 * 64];
  __shared__ float sWb[64 * 64];
  __shared__ float sBa[64];
  __shared__ float sBb[64];
  __shared__ float sT[4][16 * 68];

  int tid = threadIdx.x;
  for (int i = tid; i < 64 * 64; i += 128) { sWa[i] = W2a[i]; sWb[i] = W2b[i]; }
  if (tid < 64) { sBa[tid] = b2a[tid]; sBb[tid] = b2b[tid]; }
  __syncthreads();

  int wave = tid >> 5;
  int lane = tid & 31;
  int half = lane >> 4;
  int m    = lane & 15;
  int tile = blockIdx.x * 4 + wave;
  if (tile >= N_TILES) return;
  int node0 = tile * 16;
  float* myT = &sT[wave][0];

  // Stage A = h1 + agg64 (16x64 tile) to LDS, coalesced.
  for (int i = lane; i < 16 * 64; i += 32) {
    int r = i >> 6, c = i & 63;
    int node = node0 + r;
    myT[r * 68 + c] = h1[node * 64 + c] + agg64[node * 64 + c];
  }
  // Same-wave LDS ops are in-order (DScnt); compiler inserts waits on reuse.

  // ---- GEMM1: A[16x64] @ W2a[64x64] ----
  v8f acc[4] = {};
  for (int ks = 0; ks < 16; ++ks) {
    int kk = ks * 4 + 2 * half;
    v2f a;
    a.x = myT[m * 68 + kk];
    a.y = myT[m * 68 + kk + 1];
    #pragma unroll
    for (int nt = 0; nt < 4; ++nt) {
      v2f b;
      b.x = sWa[kk * 64 + nt * 16 + m];
      b.y = sWa[(kk + 1) * 64 + nt * 16 + m];
      acc[nt] = __builtin_amdgcn_wmma_f32_16x16x4_f32(
          false, a, false, b, (short)0, acc[nt], false, false);
    }
  }

  // bias + relu -> overwrite myT with t (all A reads precede these, in-order)
  #pragma unroll
  for (int nt = 0; nt < 4; ++nt) {
    float bias = sBa[nt * 16 + m];
    #pragma unroll
    for (int r = 0; r < 8; ++r) {
      float v = acc[nt][r] + bias;
      myT[(r + 8 * half) * 68 + nt * 16 + m] = v > 0.0f ? v : 0.0f;
    }
  }

  // ---- GEMM2: t[16x64] @ W2b[64x64] ----
  v8f acc2[4] = {};
  for (int ks = 0; ks < 16; ++ks) {
    int kk = ks * 4 + 2 * half;
    v2f a2;
    a2.x = myT[m * 68 + kk];
    a2.y = myT[m * 68 + kk + 1];
    #pragma unroll
    for (int nt = 0; nt < 4; ++nt) {
      v2f b;
      b.x = sWb[kk * 64 + nt * 16 + m];
      b.y = sWb[(kk + 1) * 64 + nt * 16 + m];
      acc2[nt] = __builtin_amdgcn_wmma_f32_16x16x4_f32(
          false, a2, false, b, (short)0, acc2[nt], false, false);
    }
  }

  #pragma unroll
  for (int nt = 0; nt < 4; ++nt) {
    float bias = sBb[nt * 16 + m];
    #pragma unroll
    for (int r = 0; r < 8; ++r) {
      float v = acc2[nt][r] + bias;
      h2[(node0 + r + 8 * half) * 64 + nt * 16 + m] = v > 0.0f ? v : 0.0f;
    }
  }
}

__global__ void pool_add_kernel(const float* __restrict__ h,
                                const int* __restrict__ batch,
                                float* __restrict__ sums) {
  long long gid = (long long)blockIdx.x * blockDim.x + threadIdx.x;
  int n = (int)(gid >> 6);
  int f = (int)(gid & 63);
  if (n >= N_NODES) return;
  int g = batch[n];
  atomicAdd(&sums[g * 64 + f], h[n * 64 + f]);
}

__global__ void count_kernel(const int* __restrict__ batch,
                             float* __restrict__ cnt) {
  int n = blockIdx.x * blockDim.x + threadIdx.x;
  if (n >= N_NODES) return;
  atomicAdd(&cnt[batch[n]], 1.0f);
}

__global__ void finalize_kernel(const float* __restrict__ sums,
                                const float* __restrict__ cnt,
                                float* __restrict__ out) {
  int i = blockIdx.x * blockDim.x + threadIdx.x;
  if (i >= N_GRAPHS * 64) return;
  float c = cnt[i >> 6];
  out[i] = sums[i] / fmaxf(c, 1.0f);
}

extern "C" void kernel_launch(void* const* d_in, const int* in_sizes, int n_in,
                              void* d_out, int out_size, void* d_ws, size_t ws_size,
                              hipStream_t stream) {
  const float* x     = (const float*)d_in[0];
  const int*   ei    = (const int*)d_in[1];   // [2, E]: src then dst
  const int*   batch = (const int*)d_in[2];
  const float* W1a   = (const float*)d_in[3];
  const float* b1a   = (const float*)d_in[4];
  const float* W1b   = (const float*)d_in[5];
  const float* b1b   = (const float*)d_in[6];
  const float* W2a   = (const float*)d_in[7];
  const float* b2a   = (const float*)d_in[8];
  const float* W2b   = (const float*)d_in[9];
  const float* b2b   = (const float*)d_in[10];
  float* out = (float*)d_out;

  // Workspace layout (floats): agg3 | h1 (reused as h2 in-place) | agg64 | sums | cnt
  float* ws    = (float*)d_ws;
  float* agg3  = ws;                              // 300000
  float* h1    = ws + 300000;                     // 6400000 (h2 aliases this)
  float* agg64 = ws + 300000 + 6400000;           // 6400000
  float* sums  = ws + 300000 + 12800000;          // 8192
  // cnt = sums + 8192                            // 128  (zeroed with sums)

  dim3 b256(256);
  zero_kernel<<<(300000 + 255) / 256, b256, 0, stream>>>(agg3, 300000);
  zero_kernel<<<(6400000 + 255) / 256, b256, 0, stream>>>(agg64, 6400000);
  zero_kernel<<<(8320 + 255) / 256, b256, 0, stream>>>(sums, 8320);

  scatter3_kernel<<<(N_EDGES + 255) / 256, b256, 0, stream>>>(x, ei, agg3);
  mlp1_kernel<<<(N_TILES + 3) / 4, dim3(128), 0, stream>>>(
      x, agg3, W1a, b1a, W1b, b1b, h1);
  scatter64_kernel<<<(N_EDGES * 64) / 256, b256, 0, stream>>>(h1, ei, agg64);
  mlp2_kernel<<<(N_TILES + 3) / 4, dim3(128), 0, stream>>>(
      h1, agg64, W2a, b2a, W2b, b2b, h1 /* in-place h2 */);
  pool_add_kernel<<<(N_NODES * 64 + 255) / 256, b256, 0, stream>>>(h1, batch, sums);
  count_kernel<<<(N_NODES + 255) / 256, b256, 0, stream>>>(batch, sums + 8192);
  finalize_kernel<<<(N_GRAPHS * 64 + 255) / 256, b256, 0, stream>>>(
      sums, sums + 8192, out);
}